// SoftCrossEntropyLoss_83476984365621
// MI455X (gfx1250) — compile-verified
//
#include <hip/hip_runtime.h>

typedef __attribute__((ext_vector_type(2))) float v2f;
typedef __attribute__((ext_vector_type(8))) float v8f;
typedef __attribute__((ext_vector_type(4))) unsigned int u32x4;
typedef __attribute__((ext_vector_type(4))) int i32x4;
typedef __attribute__((ext_vector_type(8))) int i32x8;

#define KDIM 256
#define ROWS_PER_WAVE 16
#define WAVES_PER_BLOCK 4
#define ROWS_PER_BLOCK 64          // 4 waves * 16 rows
#define LDS_ROW_F 260              // 256 floats + 4 floats TDM pad -> conflict-free banks
#define TILE_F (ROWS_PER_WAVE * LDS_ROW_F)   // 4160 floats per staged tile

// ---------------------------------------------------------------------------
// TDM: DMA one 16-row x 256-col f32 tile (row-major, stride 256 floats in
// memory) into LDS at lds_byte_off, inserting 4 DWORDs of padding after every
// 256 DWORDs (one row) => LDS row stride = 260 floats.
// D# layout per CDNA5 ISA.
// ---------------------------------------------------------------------------
__device__ __forceinline__ void tdm_load_tile16x256(const float* gptr,
                                                    unsigned lds_byte_off) {
    unsigned long long ga = (unsigned long long)(uintptr_t)gptr;

    u32x4 g0;
    g0.x = 1u;                                   // count=1 (valid user descriptor)
    g0.y = lds_byte_off;                         // lds_addr [63:32]
    g0.z = (unsigned)(ga & 0xFFFFFFFFu);         // global_addr [95:64]
    g0.w = (unsigned)((ga >> 32) & 0x01FFFFFFu)  // global_addr [120:96]
         | (2u << 30);                           // type=2 ("image") [127:126]

    i32x8 g1;
    g1[0] = (int)((2u << 16)      // data_size = 2 -> 4 bytes
                | (1u << 20)      // pad_enable
                | (7u << 22)      // pad_interval = 7 -> every 256 DWORDs
                | (3u << 25));    // pad_amount   = 3 -> 4 DWORDs
    g1[1] = (int)(256u << 16);    // tensor_dim0[15:0]=256 (bits 63:48)
    g1[2] = (int)(16u << 16);     // tensor_dim0 hi=0 | tensor_dim1[15:0]=16
    g1[3] = (int)(256u << 16);    // tensor_dim1 hi=0 | tile_dim0=256
    g1[4] = 16;                   // tile_dim1=16, tile_dim2=0
    g1[5] = 256;                  // tensor_dim0_stride low32 = 256 elements
    g1[6] = 0;                    // stride hi | tensor_dim1_stride lo (unused, 2D)
    g1[7] = 0;

    i32x4 z4 = {0, 0, 0, 0};
#if __clang_major__ >= 23
    i32x8 z8 = {0, 0, 0, 0, 0, 0, 0, 0};
    __builtin_amdgcn_tensor_load_to_lds(g0, g1, z4, z4, z8, 0);
#else
    __builtin_amdgcn_tensor_load_to_lds(g0, g1, z4, z4, 0);
#endif
}

// ---------------------------------------------------------------------------
// Main kernel: each wave owns a 16x256 tile of x and t, staged by TDM.
// Lane mapping matches V_WMMA_F32_16X16X4_F32 A-matrix layout (ISA 7.12.2):
//   lane r (r<16): A[r, k] for k = 4c+{0,1};  lane r+16: k = 4c+{2,3}
// B = w[4c..4c+3] broadcast across columns (one ds_load_b64 from lds_w).
// WMMA accumulates s2 = sum t*w and s1 = sum t*w*x over all 64 K-chunks.
// loss_row = (m_row + log(sum exp(x - m_row))) * s2_row - s1_row.
// ---------------------------------------------------------------------------
__global__ __launch_bounds__(128) void sce_main(const float* __restrict__ x,
                                                const float* __restrict__ t,
                                                const float* __restrict__ w,
                                                float* __restrict__ partials) {
    __shared__ float lds_w[KDIM];
    __shared__ float xbuf[WAVES_PER_BLOCK][TILE_F];
    __shared__ float tbuf[WAVES_PER_BLOCK][TILE_F];
    __shared__ float wave_tot[WAVES_PER_BLOCK];

    const int tid = threadIdx.x;
    lds_w[tid]       = w[tid];
    lds_w[tid + 128] = w[tid + 128];

    const int wave = tid >> 5;
    const int lane = tid & 31;
    const int half = lane >> 4;    // 0: k%4 in {0,1}, 1: k%4 in {2,3}
    const int r    = lane & 15;    // row within tile
    const long long row_base = (long long)blockIdx.x * ROWS_PER_BLOCK
                             + (long long)wave * ROWS_PER_WAVE;

    // Kick off the two DMAs for this wave's tile (wave-level ops, EXEC ignored).
    tdm_load_tile16x256(x + row_base * KDIM,
                        (unsigned)(uintptr_t)&xbuf[wave][0]);
    tdm_load_tile16x256(t + row_base * KDIM,
                        (unsigned)(uintptr_t)&tbuf[wave][0]);

    __syncthreads();                       // lds_w visible to all waves
    __builtin_amdgcn_s_wait_tensorcnt(0);  // this wave's tiles are in LDS
    asm volatile("" ::: "memory");         // TDM wrote LDS behind compiler's back

    const float* xl = &xbuf[wave][r * LDS_ROW_F + 2 * half];
    const float* tl = &tbuf[wave][r * LDS_ROW_F + 2 * half];

    // ---- Pass 1: per-row max from LDS (lane pair covers full row) ----
    float m = -3.402823466e38f;
#pragma unroll 8
    for (int c = 0; c < KDIM / 4; ++c) {
        v2f xv = *(const v2f*)(xl + 4 * c);
        m = fmaxf(m, fmaxf(xv.x, xv.y));
    }
    m = fmaxf(m, __shfl_xor(m, 16, 32));   // lanes r and r+16 hold m_row

    // ---- Pass 2: sumexp (VALU) + s1,s2 (WMMA vs broadcast-w), all from LDS --
    float sum_e = 0.0f;
    v8f d_s1 = {0.f, 0.f, 0.f, 0.f, 0.f, 0.f, 0.f, 0.f};  // sum t*w*x
    v8f d_s2 = {0.f, 0.f, 0.f, 0.f, 0.f, 0.f, 0.f, 0.f};  // sum t*w
#pragma unroll 4
    for (int c = 0; c < KDIM / 4; ++c) {
        v2f xv = *(const v2f*)(xl + 4 * c);
        v2f tv = *(const v2f*)(tl + 4 * c);
        v2f wb = *(const v2f*)(&lds_w[4 * c + 2 * half]);

        sum_e += __expf(xv.x - m) + __expf(xv.y - m);

        v2f a1;
        a1.x = tv.x * xv.x;
        a1.y = tv.y * xv.y;

        // (neg_a, A, neg_b, B, c_mod, C, reuse_a, reuse_b)
        d_s2 = __builtin_amdgcn_wmma_f32_16x16x4_f32(false, tv, false, wb,
                                                     (short)0, d_s2, false, false);
        d_s1 = __builtin_amdgcn_wmma_f32_16x16x4_f32(false, a1, false, wb,
                                                     (short)0, d_s1, false, false);
    }

    sum_e += __shfl_xor(sum_e, 16, 32);          // full-row sumexp
    const float lse = m + __logf(sum_e);         // full log-sum-exp (lanes r, r+16)

    // D layout: VGPR j = row j (lanes 0-15) / row j+8 (lanes 16-31); N-invariant.
    float acc = 0.0f;
#pragma unroll
    for (int j = 0; j < 8; ++j) {
        const int src = half ? (j + 8) : j;      // lane holding lse for that row
        const float lj = __shfl(lse, src, 32);
        acc += lj * d_s2[j] - d_s1[j];           // uniform within each lane half
    }
    acc += __shfl_xor(acc, 16, 32);              // rows 0-7 + rows 8-15

    if (lane == 0) wave_tot[wave] = acc;
    __syncthreads();
    if (tid == 0) {
        float b = 0.0f;
#pragma unroll
        for (int i = 0; i < WAVES_PER_BLOCK; ++i) b += wave_tot[i];
        partials[blockIdx.x] = b;
    }
}

// Deterministic final reduction: fixed-order strided sums + fixed LDS tree.
__global__ __launch_bounds__(256) void sce_reduce(const float* __restrict__ partials,
                                                  float* __restrict__ out,
                                                  int nblocks, float inv_n) {
    __shared__ float s[256];
    float a = 0.0f;
    for (int i = threadIdx.x; i < nblocks; i += 256) a += partials[i];
    s[threadIdx.x] = a;
    __syncthreads();
    for (int st = 128; st > 0; st >>= 1) {
        if ((int)threadIdx.x < st) s[threadIdx.x] += s[threadIdx.x + st];
        __syncthreads();
    }
    if (threadIdx.x == 0) out[0] = s[0] * inv_n;
}

extern "C" void kernel_launch(void* const* d_in, const int* in_sizes, int n_in,
                              void* d_out, int out_size, void* d_ws, size_t ws_size,
                              hipStream_t stream) {
    const float* x = (const float*)d_in[0];   // logits [N, K]
    const float* t = (const float*)d_in[1];   // target [N, K]
    const float* w = (const float*)d_in[2];   // weight [K]
    float* out = (float*)d_out;

    const int k = in_sizes[2];                        // 256
    const long long n = (long long)in_sizes[0] / k;   // 524288
    const int nblocks = (int)(n / ROWS_PER_BLOCK);    // 8192 (divides evenly)
    float* partials = (float*)d_ws;

    sce_main<<<nblocks, 128, 0, stream>>>(x, t, w, partials);
    sce_reduce<<<1, 256, 0, stream>>>(partials, out, nblocks, 1.0f / (float)n);
}